// POSTree_21182778704987
// MI455X (gfx1250) — compile-verified
//
#include <hip/hip_runtime.h>
#include <hip/hip_bf16.h>
#include <stdint.h>

// Problem constants (match reference).
#define DD 128          // embedding dim
#define TT 17           // number of tags
#define NN 87381        // nodes in complete 4-ary tree of depth 9
#define WSTRIDE 132     // LDS row stride (pad 4 floats -> conflict-free WMMA frag reads)
#define LDS_FLOATS (DD * WSTRIDE + 16 * WSTRIDE)   // W tile + A tile = 19008 floats (76,032 B)

typedef __attribute__((ext_vector_type(2))) float v2f;
typedef __attribute__((ext_vector_type(8))) float v8f;

// ---------------------------------------------------------------------------
// Phase 0: counting sort of node indices by tag (17 buckets).
// ---------------------------------------------------------------------------
__global__ void zero_counts_kernel(int* __restrict__ counts) {
  if (threadIdx.x < TT) counts[threadIdx.x] = 0;
}

__global__ void hist_kernel(const int* __restrict__ tag, int* __restrict__ counts, int n) {
  __shared__ int lc[TT];
  if (threadIdx.x < TT) lc[threadIdx.x] = 0;
  __syncthreads();
  int i = blockIdx.x * blockDim.x + threadIdx.x;
  if (i < n) atomicAdd(&lc[tag[i]], 1);
  __syncthreads();
  if (threadIdx.x < TT) atomicAdd(&counts[threadIdx.x], lc[threadIdx.x]);
}

__global__ void scan_kernel(const int* __restrict__ counts, int* __restrict__ starts,
                            int* __restrict__ cursor) {
  if (threadIdx.x == 0) {
    int acc = 0;
    for (int t = 0; t < TT; ++t) { starts[t] = acc; cursor[t] = acc; acc += counts[t]; }
  }
}

__global__ void scatter_kernel(const int* __restrict__ tag, int* __restrict__ cursor,
                               int* __restrict__ bucket, int n) {
  int i = blockIdx.x * blockDim.x + threadIdx.x;
  if (i < n) {
    int pos = atomicAdd(&cursor[tag[i]], 1);
    bucket[pos] = i;
  }
}

// ---------------------------------------------------------------------------
// Phase 1: per-tag GEMM with fp32 WMMA.
//   Block = 256 threads (8 wave32). blockIdx.y = tag, blockIdx.x = 16-row tile
//   of that tag's bucket. Wave w computes output columns [16w, 16w+16).
//   y[m][n] = sum_k A[m][k] * W[t][n][k]  (A = gathered emb rows), + bias, relu.
// ---------------------------------------------------------------------------
__global__ void gemm_tag_kernel(const float* __restrict__ emb,
                                const float* __restrict__ W,
                                const float* __restrict__ bias,
                                const int*   __restrict__ bucket,
                                const int*   __restrict__ starts,
                                const int*   __restrict__ counts,
                                float*       __restrict__ x) {
  extern __shared__ float smem[];
  float* Wl = smem;                 // [128][WSTRIDE]
  float* Al = smem + DD * WSTRIDE;  // [16][WSTRIDE]

  const int t    = blockIdx.y;
  const int tile = blockIdx.x;
  const int cnt  = counts[t];
  if (tile * 16 >= cnt) return;                 // empty tile for this tag
  const int base = starts[t] + tile * 16;
  const int tid  = threadIdx.x;

  // Stage W[t] (row-major [out=128][in=128]) into padded LDS.
  const float* Wt = W + (size_t)t * DD * DD;
  for (int i = tid; i < DD * DD; i += 256) {
    int r = i >> 7, c = i & 127;
    Wl[r * WSTRIDE + c] = Wt[i];
  }
  // Stage 16 gathered emb rows (zero-pad invalid rows).
  {
    int r  = tid >> 4;            // 0..15
    int c0 = (tid & 15) * 8;      // 8 floats per thread
    bool ok = (tile * 16 + r) < cnt;
    int node = ok ? bucket[base + r] : 0;
    const float* src = emb + (size_t)node * DD + c0;
    float* dst = Al + r * WSTRIDE + c0;
#pragma unroll
    for (int j = 0; j < 8; ++j) dst[j] = ok ? src[j] : 0.0f;
  }
  __syncthreads();

  const int wave = tid >> 5;          // 0..7
  const int lane = tid & 31;
  const int n0   = wave * 16;
  const int lm   = lane & 15;         // row-within-halfwave
  const int lh   = lane >> 4;         // half-wave select (K offset +2)

  // A frag: lane holds A[lm][k+2*lh .. +1]; B frag: W[t][n0+lm][k+2*lh .. +1].
  const float* arow = Al + lm * WSTRIDE + 2 * lh;
  const float* brow = Wl + (n0 + lm) * WSTRIDE + 2 * lh;

  v8f acc = {};
#pragma unroll
  for (int k = 0; k < DD; k += 4) {
    v2f a;  a.x = arow[k];  a.y = arow[k + 1];
    v2f b;  b.x = brow[k];  b.y = brow[k + 1];
    acc = __builtin_amdgcn_wmma_f32_16x16x4_f32(
        /*neg_a=*/false, a, /*neg_b=*/false, b,
        /*c_mod=*/(short)0, acc, /*reuse_a=*/false, /*reuse_b=*/false);
  }

  // C/D layout: VGPR v, lanes 0-15 -> M=v, lanes 16-31 -> M=v+8; N = n0 + lm.
  const int   n  = n0 + lm;
  const float bv = bias[t * DD + n];
#pragma unroll
  for (int v = 0; v < 8; ++v) {
    int m = v + 8 * lh;
    if (tile * 16 + m < cnt) {
      int node = bucket[base + m];
      float val = acc[v] + bv;
      x[(size_t)node * DD + n] = val > 0.0f ? val : 0.0f;
    }
  }
}

// ---------------------------------------------------------------------------
// Phase 2: level-wise tree max. Parents at [pstart, pstart+pcount);
// children of p are 4p+1..4p+4.
// ---------------------------------------------------------------------------
__global__ void tree_reduce_kernel(float* __restrict__ x, int pstart, int pcount) {
  int idx = blockIdx.x * blockDim.x + threadIdx.x;
  if (idx >= pcount * DD) return;
  int p = pstart + (idx >> 7);
  int d = idx & 127;
  size_t row = (size_t)p * DD + d;
  float m = x[row];
  size_t cbase = ((size_t)4 * p + 1) * DD + d;
#pragma unroll
  for (int c = 0; c < 4; ++c) {
    float v = x[cbase + (size_t)c * DD];
    m = v > m ? v : m;
  }
  x[row] = m;
}

__global__ void copy_root_kernel(const float* __restrict__ x, float* __restrict__ out) {
  if (threadIdx.x < DD) out[threadIdx.x] = x[threadIdx.x];
}

// ---------------------------------------------------------------------------
extern "C" void kernel_launch(void* const* d_in, const int* in_sizes, int n_in,
                              void* d_out, int out_size, void* d_ws, size_t ws_size,
                              hipStream_t stream) {
  (void)in_sizes; (void)n_in; (void)out_size;
  const float* emb  = (const float*)d_in[0];   // [N, D]
  const float* W    = (const float*)d_in[1];   // [T, D, D]
  const float* bias = (const float*)d_in[2];   // [T, D]
  const int*   tag  = (const int*)d_in[3];     // [N] int32
  // d_in[4] (parent) / d_in[5] (depth) unused: heap layout is implicit.
  float* out = (float*)d_out;                  // [1, D] fp32

  // Workspace layout (all 256-byte aligned).
  char*  ws          = (char*)d_ws;
  size_t xBytes      = (size_t)NN * DD * sizeof(float);          // 44,739,072
  size_t bucketOff   = xBytes;
  size_t bucketBytes = (((size_t)NN * sizeof(int)) + 255) & ~(size_t)255;
  size_t metaOff     = bucketOff + bucketBytes;
  size_t needed      = metaOff + 3 * TT * sizeof(int) + 256;
  if (ws_size < needed) return;

  float* x      = (float*)ws;
  int*   bucket = (int*)(ws + bucketOff);
  int*   counts = (int*)(ws + metaOff);
  int*   starts = counts + TT;
  int*   cursor = starts + TT;

  // Phase 0: bucket nodes by tag.
  zero_counts_kernel<<<1, 32, 0, stream>>>(counts);
  int nb = (NN + 255) / 256;
  hist_kernel<<<nb, 256, 0, stream>>>(tag, counts, NN);
  scan_kernel<<<1, 32, 0, stream>>>(counts, starts, cursor);
  scatter_kernel<<<nb, 256, 0, stream>>>(tag, cursor, bucket, NN);

  // Phase 1: per-tag WMMA GEMM (over-launched fixed grid, empty tiles exit).
  dim3 grid((NN + 15) / 16, TT);
  gemm_tag_kernel<<<grid, 256, LDS_FLOATS * sizeof(float), stream>>>(
      emb, W, bias, bucket, starts, counts, x);

  // Phase 2: bottom-up tree max, deepest level first.
  for (int lvl = 8; lvl >= 1; --lvl) {
    int pcount = 1 << (2 * (lvl - 1));        // 4^(lvl-1) parents
    int pstart = (pcount - 1) / 3;            // (4^(lvl-1)-1)/3
    int tot = pcount * DD;
    tree_reduce_kernel<<<(tot + 255) / 256, 256, 0, stream>>>(x, pstart, pcount);
  }

  copy_root_kernel<<<1, DD, 0, stream>>>(x, out);
}